// SCNet_77146202571252
// MI455X (gfx1250) — compile-verified
//
#include <hip/hip_runtime.h>
#include <hip/hip_bf16.h>

typedef __attribute__((ext_vector_type(2))) float v2f;
typedef __attribute__((ext_vector_type(8))) float v8f;
typedef __attribute__((ext_vector_type(4))) unsigned int u32x4;
typedef __attribute__((ext_vector_type(8))) int i32x8;
typedef __attribute__((ext_vector_type(4))) int i32x4;

#define WMMA_F32X4(a, b, c) \
  __builtin_amdgcn_wmma_f32_16x16x4_f32(false, (a), false, (b), (short)0, (c), false, false)

static __device__ __forceinline__ int imin(int a, int b) { return a < b ? a : b; }

// ---------------------------------------------------------------------------
// Tensor Data Mover: 2D tile (rows x elems, fp32) global -> LDS (contiguous).
// D# layout per CDNA5 ISA. 6-arg builtin form.
static __device__ __forceinline__ void tdm_load_2d(unsigned lds_byte,
                                                   const void* gptr,
                                                   unsigned tile_d0,   // elems/row
                                                   unsigned tile_d1,   // rows
                                                   unsigned tens_d0,   // valid elems
                                                   unsigned tens_d1,   // valid rows
                                                   unsigned long long stride0) {
  unsigned long long ga = (unsigned long long)(size_t)gptr;
  u32x4 g0;
  g0[0] = 1u;                                            // count=1, user descriptor
  g0[1] = lds_byte;                                      // lds_addr (bytes)
  g0[2] = (unsigned)(ga & 0xFFFFFFFFull);                // global_addr[31:0]
  g0[3] = (unsigned)((ga >> 32) & 0x01FFFFFFull) | (2u << 30);  // addr[56:32], type=2
  i32x8 g1;
  g1[0] = (int)(2u << 16);                               // data_size=4B, mask=0
  g1[1] = (int)((tens_d0 & 0xFFFFu) << 16);              // tensor_dim0[15:0]
  g1[2] = (int)(((tens_d0 >> 16) & 0xFFFFu) | ((tens_d1 & 0xFFFFu) << 16));
  g1[3] = (int)(((tens_d1 >> 16) & 0xFFFFu) | ((tile_d0 & 0xFFFFu) << 16));
  g1[4] = (int)(tile_d1 & 0xFFFFu);                      // tile_dim1, tile_dim2=0
  g1[5] = (int)(unsigned)(stride0 & 0xFFFFFFFFull);      // tensor_dim0_stride lo32
  g1[6] = (int)(unsigned)((stride0 >> 32) & 0xFFFFull);  // stride hi16, dim1_stride=0
  g1[7] = 0;
  i32x4 z4 = {0, 0, 0, 0};
  i32x8 z8 = {0, 0, 0, 0, 0, 0, 0, 0};
  __builtin_amdgcn_tensor_load_to_lds(g0, g1, z4, z4, z8, 0);
}

static __device__ __forceinline__ unsigned lds_off(const void* p) {
  return (unsigned)(size_t)p;  // LDS flat addr: low 32 bits are the LDS offset
}

// ---- workspace layout (float offsets) -------------------------------------
static constexpr size_t OFF_E    = 0;         // 128*512 = 65536
static constexpr size_t OFF_WT   = 65536;     // repacked conv w = 537600
static constexpr size_t OFF_UST  = 603136;    // 128*64 = 8192
static constexpr size_t OFF_Q    = 611328;    // 64
static constexpr size_t OFF_S    = 611392;    // 128
static constexpr size_t OFF_CC   = 611520;    // 8
static constexpr size_t OFF_UC1  = 620000;    // [u_mm | u_mlp] 128*10000
static constexpr size_t OFF_UC2  = 1900000;   // [u_sc | u_fic] 128*10000
static constexpr size_t OFF_UMMF = 3180000;   // 128*5000 ; total ~14.6 MB

static constexpr size_t WT_B0 = OFF_WT;
static constexpr size_t WT_B1 = OFF_WT + 76800;
static constexpr size_t WT_B2 = OFF_WT + 192000;
static constexpr size_t WT_B3 = OFF_WT + 345600;

// ---------------------------------------------------------------------------
__global__ __launch_bounds__(256) void zero_kernel(float* p, int n) {
  int i = blockIdx.x * 256 + threadIdx.x;
  if (i < n) p[i] = 0.0f;
}

// Repack conv weights (K,C,ws) -> (K, ws, C): contraction r = j*300 + c
__global__ __launch_bounds__(256) void repack_w_kernel(const float* __restrict__ src,
                                                       float* __restrict__ dst,
                                                       int ws, int total) {
  int i = blockIdx.x * 256 + threadIdx.x;
  if (i < total) {
    int cw = 300 * ws;
    int k = i / cw;
    int rem = i - k * cw;
    int c = rem / ws;
    int j = rem - c * ws;
    dst[k * cw + j * 300 + c] = src[i];
  }
}

// ---------------------------------------------------------------------------
// TextCNN branch: conv1d(VALID)+relu+global max via V_WMMA_F32_16X16X4_F32.
// x rows staged by TDM into a 6-slot circular LDS buffer xs[slot][16 b][300 c],
// prefetch depth 1 (issue row l+WSV while computing position l).
template <int WSV, int BR>
__global__ __launch_bounds__(256) void conv_kernel(const float* __restrict__ x,
                                                   const float* __restrict__ wT,
                                                   const float* __restrict__ bias,
                                                   float* __restrict__ e) {
  extern __shared__ float xs[];  // 6*4800 floats = 115200 B
  const int tid  = threadIdx.x;
  const int wave = tid >> 5;
  const int lane = tid & 31;
  const int lh   = lane >> 4;
  const int mn   = lane & 15;
  const int b0   = blockIdx.x * 16;
  const int P    = 200 - WSV + 1;
  const int l0   = blockIdx.y * 16;
  const int l1   = imin(l0 + 16, P);
  if (l0 >= P) return;
  const int last_row = l1 - 1 + WSV - 1;

  v8f vmax = {0.f, 0.f, 0.f, 0.f, 0.f, 0.f, 0.f, 0.f};  // relu floor
  const float* wbase = wT + (size_t)(wave * 16 + mn) * (300 * WSV) + 2 * lh;
  const float* xblk = x + (size_t)b0 * 60000;  // batch stride 200*300

  if (wave == 0) {
    // preload the WSV rows needed for l = l0
    for (int row = l0; row < l0 + WSV; ++row) {
      int slot = row % 6;
      tdm_load_2d(lds_off(&xs[slot * 4800]), xblk + (size_t)row * 300,
                  300u, 16u, 300u, 16u, 60000ull);
    }
  }

  for (int l = l0; l < l1; ++l) {
    if (wave == 0) {
      int rown = l + WSV;  // prefetch one row ahead
      if (rown <= last_row) {
        int slot = rown % 6;
        tdm_load_2d(lds_off(&xs[slot * 4800]), xblk + (size_t)rown * 300,
                    300u, 16u, 300u, 16u, 60000ull);
        __builtin_amdgcn_s_wait_tensorcnt(1);
      } else {
        __builtin_amdgcn_s_wait_tensorcnt(0);
      }
    }
    __syncthreads();  // rows l..l+WSV-1 visible to all waves

    v8f d = {0.f, 0.f, 0.f, 0.f, 0.f, 0.f, 0.f, 0.f};
    for (int j = 0; j < WSV; ++j) {
      int slot = (l + j) % 6;
      const float* xrow = &xs[slot * 4800 + mn * 300 + 2 * lh];
      const float* wrow = wbase + j * 300;
#pragma unroll 5
      for (int c0 = 0; c0 < 300; c0 += 4) {
        v2f av = *(const v2f*)(xrow + c0);   // ds_load_b64 (A: x window)
        v2f bv = *(const v2f*)(wrow + c0);   // global_load_b64 (B: L2-resident w)
        d = WMMA_F32X4(av, bv, d);
      }
    }
    float bval = bias[wave * 16 + mn];
#pragma unroll
    for (int v = 0; v < 8; ++v) {
      float r = d[v] + bval;
      if (r > vmax[v]) vmax[v] = r;
    }
    __syncthreads();  // all reads done before the slot is DMA-overwritten
  }

  // cross-chunk max reduction: relu output >= 0 -> float bits are monotonic
#pragma unroll
  for (int v = 0; v < 8; ++v) {
    int b = b0 + v + 8 * lh;
    int k = wave * 16 + mn;
    atomicMax((unsigned int*)&e[(size_t)b * 512 + k * 4 + BR], __float_as_uint(vmax[v]));
  }
}

// ---------------------------------------------------------------------------
// u_sc_trans = e@fic_fc_w^T + b ; rank-1 MLP factors g,q,c ; s[b]
__global__ __launch_bounds__(256) void small_kernel(const float* __restrict__ e,
                                                    const float* __restrict__ ffw,
                                                    const float* __restrict__ ffb,
                                                    const float* __restrict__ w1,
                                                    const float* __restrict__ b1,
                                                    const float* __restrict__ w2,
                                                    const float* __restrict__ b2,
                                                    float* __restrict__ ust,
                                                    float* __restrict__ q_ws,
                                                    float* __restrict__ s_ws,
                                                    float* __restrict__ cc_ws) {
  __shared__ float s_ust[128 * 64];
  __shared__ float sg[64];
  const int tid = threadIdx.x;

  for (int idx = tid; idx < 128 * 64; idx += 256) {
    int b = idx >> 6, f = idx & 63;
    float acc = ffb[f];
    const float* ep = e + (size_t)b * 512;
    const float* wp = ffw + (size_t)f * 512;
    for (int r = 0; r < 512; ++r) acc += ep[r] * wp[r];
    s_ust[idx] = acc;
    ust[idx] = acc;
  }
  if (tid < 64) {
    float gg = 0.f, qq = 0.f;
    for (int fo = 0; fo < 64; ++fo) {
      float wv = w2[fo];
      gg += w1[fo * 128 + tid] * wv;
      qq += w1[fo * 128 + 64 + tid] * wv;
    }
    sg[tid] = gg;
    q_ws[tid] = qq;
  }
  if (tid == 64) {
    float cc = b2[0];
    for (int fo = 0; fo < 64; ++fo) cc += b1[fo] * w2[fo];
    cc_ws[0] = cc;
  }
  __syncthreads();
  if (tid < 128) {
    float sv = 0.f;
    for (int f = 0; f < 64; ++f) sv += s_ust[tid * 64 + f] * sg[f];
    s_ws[tid] = sv;
  }
}

// ---------------------------------------------------------------------------
// u_mlp[b,a] = tanh(s[b] + t[a] + c), t[a] = emb[:,a]·q -> ucat1[:, 5000+a]
__global__ __launch_bounds__(256) void tmlp_kernel(const float* __restrict__ emb,
                                                   const float* __restrict__ q_ws,
                                                   const float* __restrict__ s_ws,
                                                   const float* __restrict__ cc_ws,
                                                   float* __restrict__ uc1) {
  int a = blockIdx.x * 256 + threadIdx.x;
  if (a >= 5000) return;
  float tv = cc_ws[0];
  for (int f = 0; f < 64; ++f) tv += emb[(size_t)f * 5000 + a] * q_ws[f];
  for (int b = 0; b < 128; ++b)
    uc1[(size_t)b * 10000 + 5000 + a] = tanhf(s_ws[b] + tv);
}

// ---------------------------------------------------------------------------
// u_mm = u_sc_trans(128x64) @ emb(64x5000), WMMA -> ucat1[:, :5000]
__global__ __launch_bounds__(256) void umm_kernel(const float* __restrict__ ust,
                                                  const float* __restrict__ emb,
                                                  float* __restrict__ uc1) {
  const int tid = threadIdx.x;
  const int wave = tid >> 5, lane = tid & 31;
  const int lh = lane >> 4, mn = lane & 15;
  const int a0 = blockIdx.x * 16;
  const int acol = imin(a0 + mn, 4999);
  const float* ap = ust + (size_t)(wave * 16 + mn) * 64 + 2 * lh;
  v8f acc = {0.f, 0.f, 0.f, 0.f, 0.f, 0.f, 0.f, 0.f};
#pragma unroll
  for (int r0 = 0; r0 < 64; r0 += 4) {
    v2f av = *(const v2f*)(ap + r0);
    v2f bv;
    bv[0] = emb[(size_t)(r0 + 2 * lh) * 5000 + acol];
    bv[1] = emb[(size_t)(r0 + 2 * lh + 1) * 5000 + acol];
    acc = WMMA_F32X4(av, bv, acc);
  }
  bool ok = (a0 + mn) < 5000;
#pragma unroll
  for (int v = 0; v < 8; ++v) {
    int b = wave * 16 + v + 8 * lh;
    if (ok) uc1[(size_t)b * 10000 + a0 + mn] = acc[v];
  }
}

// ---------------------------------------------------------------------------
// out = act(A(128 x Kdim) @ W^T + bias), W (N x Kdim) row-major.
// W chunks (16 rows x kcs) DMA'd by the TDM into double-buffered LDS; chunk i+1
// is in flight while all 8 waves run WMMA on chunk i. Tail-N rows zero-filled
// by TDM OOB semantics (tensor_dim1 = valid rows).
__global__ __launch_bounds__(256) void gemm_wt_kernel(const float* __restrict__ Ain, int lda,
                                                      const float* __restrict__ W, int ldw,
                                                      const float* __restrict__ bias,
                                                      float* __restrict__ out, int ldo,
                                                      int N, int Kdim, int act) {
  __shared__ float wlds[2][16 * 256];  // 32 KB
  const int tid = threadIdx.x;
  const int wave = tid >> 5, lane = tid & 31;
  const int lh = lane >> 4, mn = lane & 15;
  const int n0 = blockIdx.x * 16;
  const int b0w = wave * 16;
  const int nvalid = imin(16, N - n0);
  const int nchunks = (Kdim + 255) >> 8;
  const float* abase = Ain + (size_t)(b0w + mn) * lda + 2 * lh;
  const float* wrow0 = W + (size_t)n0 * ldw;

  v8f acc = {0.f, 0.f, 0.f, 0.f, 0.f, 0.f, 0.f, 0.f};

  if (wave == 0) {
    int kcs0 = imin(256, Kdim);
    tdm_load_2d(lds_off(&wlds[0][0]), wrow0, (unsigned)kcs0, 16u,
                (unsigned)kcs0, (unsigned)nvalid, (unsigned long long)ldw);
  }

  for (int i = 0; i < nchunks; ++i) {
    const int k0 = i << 8;
    const int kcs = imin(256, Kdim - k0);
    if (wave == 0) {
      if (i + 1 < nchunks) {
        int k1 = (i + 1) << 8;
        int kcs1 = imin(256, Kdim - k1);
        tdm_load_2d(lds_off(&wlds[(i + 1) & 1][0]), wrow0 + k1, (unsigned)kcs1, 16u,
                    (unsigned)kcs1, (unsigned)nvalid, (unsigned long long)ldw);
        __builtin_amdgcn_s_wait_tensorcnt(1);  // chunk i done, i+1 in flight
      } else {
        __builtin_amdgcn_s_wait_tensorcnt(0);
      }
    }
    __syncthreads();

    const float* ap = abase + k0;
    const float* wl = &wlds[i & 1][mn * kcs + 2 * lh];  // TDM packs rows at kcs pitch
    if ((lane & 7) == 0 && i + 1 < nchunks)
      __builtin_prefetch(ap + 256, 0, 0);  // global_prefetch_b8: next A chunk
    const int steps = kcs >> 2;
#pragma unroll 4
    for (int st = 0; st < steps; ++st) {
      const int kk4 = st << 2;
      v2f av = *(const v2f*)(ap + kk4);   // global_load_b64 (A rows, L2-resident)
      v2f bv = *(const v2f*)(wl + kk4);   // ds_load_b64 (DMA-staged W tile)
      acc = WMMA_F32X4(av, bv, acc);
    }
    __syncthreads();  // reads done before this buffer is DMA-overwritten
  }

  const float bval = bias[imin(n0 + mn, N - 1)];
  const bool ok = (n0 + mn) < N;
#pragma unroll
  for (int v = 0; v < 8; ++v) {
    float r = acc[v] + bval;
    if (act == 1) r = tanhf(r);
    int rowg = b0w + v + 8 * lh;
    if (ok) out[(size_t)rowg * ldo + (n0 + mn)] = r;
  }
}

// ---------------------------------------------------------------------------
extern "C" void kernel_launch(void* const* d_in, const int* in_sizes, int n_in,
                              void* d_out, int out_size, void* d_ws, size_t ws_size,
                              hipStream_t stream) {
  const float* x     = (const float*)d_in[0];
  const float* cw[4] = {(const float*)d_in[1], (const float*)d_in[3],
                        (const float*)d_in[5], (const float*)d_in[7]};
  const float* cb[4] = {(const float*)d_in[2], (const float*)d_in[4],
                        (const float*)d_in[6], (const float*)d_in[8]};
  const float* sc_w  = (const float*)d_in[9];
  const float* sc_b  = (const float*)d_in[10];
  const float* ff_w  = (const float*)d_in[11];
  const float* ff_b  = (const float*)d_in[12];
  const float* emb   = (const float*)d_in[13];
  const float* w1    = (const float*)d_in[14];
  const float* b1    = (const float*)d_in[15];
  const float* w2    = (const float*)d_in[16];
  const float* b2    = (const float*)d_in[17];
  const float* fic_w = (const float*)d_in[18];
  const float* fic_b = (const float*)d_in[19];
  const float* fus_w = (const float*)d_in[20];
  const float* fus_b = (const float*)d_in[21];
  const float* tk_w  = (const float*)d_in[22];
  const float* tk_b  = (const float*)d_in[23];

  float* wsf = (float*)d_ws;
  float* out = (float*)d_out;
  float* e   = wsf + OFF_E;
  float* wT[4] = {wsf + WT_B0, wsf + WT_B1, wsf + WT_B2, wsf + WT_B3};

  // 0) e <- 0 (relu floor / atomic-max identity)
  zero_kernel<<<256, 256, 0, stream>>>(e, 65536);

  // 1) repack conv weights (K,C,ws) -> (K, ws, C)
  for (int br = 0; br < 4; ++br) {
    int ws = 2 + br;
    int tot = 128 * 300 * ws;
    repack_w_kernel<<<(tot + 255) / 256, 256, 0, stream>>>(cw[br], wT[br], ws, tot);
  }

  // 2) TextCNN branches (WMMA fp32 + TDM row staging)
  conv_kernel<2, 0><<<dim3(8, 13), 256, 115200, stream>>>(x, wT[0], cb[0], e);
  conv_kernel<3, 1><<<dim3(8, 13), 256, 115200, stream>>>(x, wT[1], cb[1], e);
  conv_kernel<4, 2><<<dim3(8, 13), 256, 115200, stream>>>(x, wT[2], cb[2], e);
  conv_kernel<5, 3><<<dim3(8, 13), 256, 115200, stream>>>(x, wT[3], cb[3], e);

  // 3) u_sc_trans, rank-1 MLP factors, s
  small_kernel<<<1, 256, 0, stream>>>(e, ff_w, ff_b, w1, b1, w2, b2,
                                      wsf + OFF_UST, wsf + OFF_Q, wsf + OFF_S, wsf + OFF_CC);

  // 4) u_mlp into ucat1[:, 5000:]
  tmlp_kernel<<<(5000 + 255) / 256, 256, 0, stream>>>(emb, wsf + OFF_Q, wsf + OFF_S,
                                                      wsf + OFF_CC, wsf + OFF_UC1);

  // 5) u_mm = ust @ emb into ucat1[:, :5000]
  umm_kernel<<<313, 256, 0, stream>>>(wsf + OFF_UST, emb, wsf + OFF_UC1);

  // 6) u_sc = e @ sc_fcl_w^T into ucat2[:, :5000]
  gemm_wt_kernel<<<313, 256, 0, stream>>>(e, 512, sc_w, 512, sc_b,
                                          wsf + OFF_UC2, 10000, 5000, 512, 0);

  // 7) u_fic = tanh(ucat1 @ fic_fcl_w^T + b) into ucat2[:, 5000:]
  gemm_wt_kernel<<<313, 256, 0, stream>>>(wsf + OFF_UC1, 10000, fic_w, 10000, fic_b,
                                          wsf + OFF_UC2 + 5000, 10000, 5000, 10000, 1);

  // 8) u_mmf = ucat2 @ fusion_w^T + b
  gemm_wt_kernel<<<313, 256, 0, stream>>>(wsf + OFF_UC2, 10000, fus_w, 10000, fus_b,
                                          wsf + OFF_UMMF, 5000, 5000, 10000, 0);

  // 9) task = u_mmf @ task_w^T + b -> d_out
  gemm_wt_kernel<<<313, 256, 0, stream>>>(wsf + OFF_UMMF, 5000, tk_w, 5000, tk_b,
                                          out, 5000, 5000, 5000, 0);
}